// BlockUngrouper_43181601194864
// MI455X (gfx1250) — compile-verified
//
#include <hip/hip_runtime.h>
#include <stdint.h>

// Problem constants (BlockUngrouper): B=16, N=16384, G=128, NG_MAX=128, D=256
#define B_      16
#define N_      16384
#define G_      128
#define NG_     128
#define D_      256
#define CHUNK   512
#define NCHUNK  (N_ / CHUNK)   // 32 chunks of tokens per batch sample

// ---------------------------------------------------------------------------
// Kernel A: per token, find the LAST block g with onehot > 0 (wave-per-token:
// one global_load_b128 per lane covers the whole 128-float row), and build a
// per-chunk per-block histogram in LDS for later rank offsets.
// ---------------------------------------------------------------------------
__global__ __launch_bounds__(256) void k_gidx_hist(const float* __restrict__ onehot,
                                                   int* __restrict__ g_idx,
                                                   int* __restrict__ counts) {
    __shared__ int hist[G_];
    const int wg   = blockIdx.x;            // wg = b * NCHUNK + c
    const int b    = wg / NCHUNK;
    const int c    = wg % NCHUNK;
    const int wave = threadIdx.x >> 5;      // 8 waves of 32 (wave32)
    const int lane = threadIdx.x & 31;

    if (threadIdx.x < G_) hist[threadIdx.x] = 0;
    __syncthreads();

    const int tok_per_wave = CHUNK / 8;     // 64 tokens per wave
    const int nbase = c * CHUNK + wave * tok_per_wave;

    for (int i = 0; i < tok_per_wave; ++i) {
        const int n = nbase + i;
        const float4* p = (const float4*)(onehot + ((size_t)b * N_ + n) * G_) + lane;
        if (i + 1 < tok_per_wave) {
            // streaming hint -> global_prefetch_b8
            __builtin_prefetch((const float4*)(onehot + ((size_t)b * N_ + n + 1) * G_) + lane, 0, 0);
        }
        const float4 v = *p;                // 16B/lane, 512B contiguous per wave
        int lg = -1;
        const int gb = lane * 4;
        if (v.x > 0.f) lg = gb + 0;
        if (v.y > 0.f) lg = gb + 1;
        if (v.z > 0.f) lg = gb + 2;
        if (v.w > 0.f) lg = gb + 3;
        // wave32 max-reduction: last claiming block wins
        #pragma unroll
        for (int off = 16; off > 0; off >>= 1) {
            const int o = __shfl_xor(lg, off, 32);
            lg = lg > o ? lg : o;
        }
        if (lane == 0) {
            g_idx[(size_t)b * N_ + n] = lg; // -1 => uncovered token
            if (lg >= 0) atomicAdd(&hist[lg], 1);
        }
    }
    __syncthreads();
    if (threadIdx.x < G_) counts[(size_t)wg * G_ + threadIdx.x] = hist[threadIdx.x];
}

// ---------------------------------------------------------------------------
// Kernel B: exclusive prefix sum over chunks for each (b, g). 2048 independent
// scans of 32 values each — trivial.
// ---------------------------------------------------------------------------
__global__ void k_scan(int* __restrict__ counts) {
    const int tid = blockIdx.x * blockDim.x + threadIdx.x;  // b*G + g
    if (tid >= B_ * G_) return;
    const int b = tid / G_;
    const int g = tid % G_;
    int sum = 0;
    for (int c = 0; c < NCHUNK; ++c) {
        const size_t idx = (size_t)(b * NCHUNK + c) * G_ + g;
        const int v = counts[idx];
        counts[idx] = sum;
        sum += v;
    }
}

// ---------------------------------------------------------------------------
// Kernel C: per chunk, compute the in-chunk ordinal rank of each token within
// its block (broadcast LDS loop, O(CHUNK) iterations), add the cross-chunk
// offset, and emit flat_idx = g*NG + min(rank, NG-1) in place over g_idx.
// ---------------------------------------------------------------------------
__global__ __launch_bounds__(256) void k_rank(int* __restrict__ g_flat,
                                              const int* __restrict__ counts) {
    __shared__ int gch[CHUNK];
    __shared__ int offs[G_];
    const int wg = blockIdx.x;                    // b * NCHUNK + c
    const size_t base = (size_t)wg * CHUNK;       // contiguous in g_flat

    const int t0 = threadIdx.x;
    const int t1 = threadIdx.x + 256;
    gch[t0] = g_flat[base + t0];
    gch[t1] = g_flat[base + t1];
    if (threadIdx.x < G_) offs[threadIdx.x] = counts[(size_t)wg * G_ + threadIdx.x];
    __syncthreads();

    const int g0 = gch[t0];
    const int g1 = gch[t1];
    int r0 = 0, r1 = 0;
    for (int tp = 0; tp < CHUNK; ++tp) {          // LDS broadcast, no conflicts
        const int gv = gch[tp];
        r0 += (int)((tp < t0) & (gv == g0));
        r1 += (int)((tp < t1) & (gv == g1));
    }
    int f0 = -1, f1 = -1;
    if (g0 >= 0) { int r = offs[g0] + r0; r = r < NG_ - 1 ? r : NG_ - 1; f0 = g0 * NG_ + r; }
    if (g1 >= 0) { int r = offs[g1] + r1; r = r < NG_ - 1 ? r : NG_ - 1; f1 = g1 * NG_ + r; }
    g_flat[base + t0] = f0;
    g_flat[base + t1] = f1;
}

// ---------------------------------------------------------------------------
// Kernel D: wave-per-token 1KB row gather through the CDNA5 async LDS path:
// global_load_async_to_lds_b128 -> s_wait_asynccnt -> store_async_from_lds.
// ---------------------------------------------------------------------------
__global__ __launch_bounds__(256) void k_gather(const float* __restrict__ feats,
                                                const int* __restrict__ flat_idx,
                                                float* __restrict__ out) {
    __shared__ float sbuf[8 * D_];                // 1KB per wave, 8 waves
    const int wave  = threadIdx.x >> 5;
    const int lane  = threadIdx.x & 31;
    const int token = blockIdx.x * 8 + wave;      // token in [0, B*N)
    const int fi    = flat_idx[token];            // uniform within the wave
    float* dst = out + (size_t)token * D_;

    if (fi < 0) {                                 // uncovered token -> zeros
        const float4 z = make_float4(0.f, 0.f, 0.f, 0.f);
        *(float4*)(dst + lane * 4)       = z;
        *(float4*)(dst + lane * 4 + 128) = z;
        return;
    }

    const int b = token >> 14;                    // N = 16384
    const float* src = feats + (size_t)b * (G_ * NG_ * D_) + (size_t)fi * D_;

    // LDS byte offsets for this wave's staging buffer (low 32 bits of the
    // generic shared-memory address are the workgroup-relative LDS offset).
    const uint32_t l0 = (uint32_t)(uintptr_t)(&sbuf[wave * D_]) + (uint32_t)lane * 16u;
    const uint32_t l1 = l0 + 512u;

    // DMA global -> LDS (2 x 16B per lane = full 1KB row per wave)
    asm volatile("global_load_async_to_lds_b128 %0, %1, off"
                 :: "v"(l0), "v"(src + lane * 4) : "memory");
    asm volatile("global_load_async_to_lds_b128 %0, %1, off"
                 :: "v"(l1), "v"(src + lane * 4 + 128) : "memory");
    asm volatile("s_wait_asynccnt 0" ::: "memory");
    // DMA LDS -> global
    asm volatile("global_store_async_from_lds_b128 %0, %1, off"
                 :: "v"(dst + lane * 4), "v"(l0) : "memory");
    asm volatile("global_store_async_from_lds_b128 %0, %1, off"
                 :: "v"(dst + lane * 4 + 128), "v"(l1) : "memory");
    asm volatile("s_wait_asynccnt 0" ::: "memory");
}

// ---------------------------------------------------------------------------
extern "C" void kernel_launch(void* const* d_in, const int* in_sizes, int n_in,
                              void* d_out, int out_size, void* d_ws, size_t ws_size,
                              hipStream_t stream) {
    const float* feats  = (const float*)d_in[0];  // [B, G, NG, D] fp32
    const float* onehot = (const float*)d_in[1];  // [B, N, G]     fp32
    float* out = (float*)d_out;                   // [B, N, D]     fp32

    // Workspace layout: g_idx/flat_idx [B*N] ints, then chunk counts [B*NCHUNK*G] ints
    int* g_idx  = (int*)d_ws;
    int* counts = g_idx + (size_t)B_ * N_;

    k_gidx_hist<<<B_ * NCHUNK, 256, 0, stream>>>(onehot, g_idx, counts);
    k_scan<<<(B_ * G_ + 255) / 256, 256, 0, stream>>>(counts);
    k_rank<<<B_ * NCHUNK, 256, 0, stream>>>(g_idx, counts);
    k_gather<<<(B_ * N_) / 8, 256, 0, stream>>>(feats, g_idx, out);
}